// GConv_31035433680984
// MI455X (gfx1250) — compile-verified
//
#include <hip/hip_runtime.h>

// =====================================================================
// SE(3)-GraphConv for MI455X (gfx1250, wave32).
// - All dense 16-wide GEMMs (radial MLPs + self-interaction MLPs) run on
//   V_WMMA_F32_16X16X4_F32: M=16 rows, N=16 features, K stepped by 4 —
//   native f32, zero conversion; r_size = J*16 tiles perfectly in N.
// - LayerNorm row-reductions use DPP16 row_ror rotations (pure VALU on
//   the 16-lane half-wave that holds one C/D row) instead of LDS
//   bpermute — keeps the WMMA->LN->WMMA dependency chain off the DS pipe.
// - The wj stream (dominant HBM traffic) is staged tile-wise into LDS,
//   via CDNA5 async global->LDS loads when the toolchain exposes them
//   (drained with s_wait_asynccnt), overlapping the WMMA layers.
// =====================================================================

typedef __attribute__((ext_vector_type(2))) float v2f;
typedef __attribute__((ext_vector_type(8))) float v8f;

#define WAVES 2      // waves (independent 16-row tiles) per block
#define RST   80     // LDS stride of the layer-3 output tile (max r_size)
#define WJMAX 2000   // 16 edges * 125 floats: largest wj tile (k=l=2)

#if __has_builtin(__builtin_amdgcn_global_load_async_to_lds_b32)
#define ASYNC_WJ 1
typedef __attribute__((address_space(1))) int* as1_i32p;
typedef __attribute__((address_space(3))) int* as3_i32p;
#endif

struct WjPtrs { const float* p[9]; };

__device__ __forceinline__ v8f wmma4(v2f a, v2f b, v8f c) {
  // D = A(16x4,f32) * B(4x16,f32) + C(16x16,f32)
  return __builtin_amdgcn_wmma_f32_16x16x4_f32(false, a, false, b,
                                               (short)0, c, false, false);
}

// ---- 16-lane row reductions via DPP16 rotations (VALU, no LDS pipe) ----
#if __has_builtin(__builtin_amdgcn_update_dpp)
template<int CTRL>
__device__ __forceinline__ float dpp_add(float x) {
  int y = __builtin_amdgcn_update_dpp(0, __builtin_bit_cast(int, x),
                                      CTRL, 0xF, 0xF, true);
  return x + __builtin_bit_cast(float, y);
}
__device__ __forceinline__ float hsum16(float x) {
  x = dpp_add<0x121>(x);   // row_ror:1
  x = dpp_add<0x122>(x);   // row_ror:2
  x = dpp_add<0x124>(x);   // row_ror:4
  x = dpp_add<0x128>(x);   // row_ror:8
  return x;                // full 16-lane row sum in every lane
}
__device__ __forceinline__ float pair_add(float x) {
  return dpp_add<0xB1>(x); // quad_perm [1,0,3,2] == xor 1
}
#else
__device__ __forceinline__ float hsum16(float x) {
  x += __shfl_xor(x, 1, 32); x += __shfl_xor(x, 2, 32);
  x += __shfl_xor(x, 4, 32); x += __shfl_xor(x, 8, 32);
  return x;
}
__device__ __forceinline__ float pair_add(float x) {
  return x + __shfl_xor(x, 1, 32);
}
#endif

// One dense layer on a 16-row tile: out[16][16] = in[16][K] @ W[16][K]^T + b,
// optional LayerNorm(+ReLU) over the 16 output features of each row.
// A frag (16x4 f32): lanes0-15 hold (K0,K1) of row=lane, lanes16-31 (K2,K3).
// B frag mirrors with N=lane&15.  D: vgpr r -> row r (+8 hi half), col=lane&15.
template<int K, bool LN>
__device__ __forceinline__ void layer16(const float* __restrict__ inT, int sin,
    const float* __restrict__ W, int ldw,
    const float* __restrict__ b, const float* __restrict__ g,
    const float* __restrict__ be,
    float* __restrict__ outT, int sout, int lane)
{
  constexpr int NS = (K + 3) / 4;
  const int n15 = lane & 15;
  const int ko  = (lane < 16) ? 0 : 2;
  v2f af[NS];
#pragma unroll
  for (int t = 0; t < NS; ++t) {          // preload ALL A frags (allows in-place)
    int k0 = t * 4 + ko;
    af[t].x = (k0     < K) ? inT[n15 * sin + k0]     : 0.f;
    af[t].y = (k0 + 1 < K) ? inT[n15 * sin + k0 + 1] : 0.f;
  }
  v8f c = {0.f,0.f,0.f,0.f,0.f,0.f,0.f,0.f};
#pragma unroll
  for (int t = 0; t < NS; ++t) {
    int k0 = t * 4 + ko;
    v2f bf;
    bf.x = (k0     < K) ? W[n15 * ldw + k0]     : 0.f;
    bf.y = (k0 + 1 < K) ? W[n15 * ldw + k0 + 1] : 0.f;
    c = wmma4(af[t], bf, c);
  }
  const int mh = (lane < 16) ? 0 : 8;
  const float bb = b[n15];
  float gg = 0.f, bebe = 0.f;
  if constexpr (LN) { gg = g[n15]; bebe = be[n15]; }
#pragma unroll
  for (int r = 0; r < 8; ++r) {
    float x = c[r] + bb;
    if constexpr (LN) {
      float mu  = hsum16(x) * 0.0625f;
      float dx  = x - mu;
      float var = hsum16(dx * dx) * 0.0625f;
      x = dx * rsqrtf(var + 1e-5f) * gg + bebe;
      x = fmaxf(x, 0.f);
    }
    outT[(mh + r) * sout + n15] = x;
  }
}

// ---------------- small helper kernels ----------------

__global__ void q_kernel(const float* __restrict__ wq,
                         const float* __restrict__ f0,
                         const float* __restrict__ f1,
                         const float* __restrict__ f2,
                         float* __restrict__ qg, int N) {
  int idx = blockIdx.x * blockDim.x + threadIdx.x;
  if (idx >= N * 36) return;
  int n = idx / 36, r = idx % 36, o = r / 9, col = r % 9;
  int k, m;
  if (col < 1)      { k = 0; m = 0; }
  else if (col < 4) { k = 1; m = col - 1; }
  else              { k = 2; m = col - 4; }
  const float* fk = (k == 0) ? f0 : ((k == 1) ? f1 : f2);
  int nl = 2 * k + 1;
  float s = 0.f;
  for (int i = 0; i < 4; ++i)
    s += wq[k * 16 + o * 4 + i] * fk[((size_t)n * 4 + i) * nl + m];
  qg[idx] = s;
}

__global__ void iv_kernel(const int* __restrict__ u, const int* __restrict__ v,
                          const float* __restrict__ dist,
                          const float* __restrict__ f0,
                          float* __restrict__ ivg, int E) {
  int e = blockIdx.x * blockDim.x + threadIdx.x;
  if (e >= E) return;
  int uu = u[e], vv = v[e];
  for (int c = 0; c < 4; ++c) ivg[e * 5 + c] = f0[uu * 4 + c] * f0[vv * 4 + c];
  ivg[e * 5 + 4] = dist[e];
}

__global__ void softmax_kernel(const float* __restrict__ dotg,
                               float* __restrict__ ag, int N, int deg) {
  int n = blockIdx.x * blockDim.x + threadIdx.x;
  if (n >= N) return;
  float m = -3.4e38f;
  for (int j = 0; j < deg; ++j) m = fmaxf(m, dotg[n * deg + j]);
  float s = 0.f;
  for (int j = 0; j < deg; ++j) s += __expf(dotg[n * deg + j] - m);
  float inv = 1.f / s;
  for (int j = 0; j < deg; ++j) ag[n * deg + j] = __expf(dotg[n * deg + j] - m) * inv;
}

__global__ void reduce_kernel(const float* __restrict__ msg,
                              float* __restrict__ out, int N, int deg) {
  int idx = blockIdx.x * blockDim.x + threadIdx.x;
  if (idx >= N * 36) return;
  int n = idx / 36, r = idx % 36;
  float s = 0.f;
  for (int j = 0; j < deg; ++j) s += msg[(size_t)(n * deg + j) * 36 + r];
  out[idx] = s;   // full overwrite: also initializes d_out
}

// ---------------- fused edge kernel (keys: s=1, values: s=0) ----------------
// One wave owns 16 edges. Per (k,l): stage the wj tile into LDS (async if
// available, overlapping the WMMAs), run 3 WMMA layers -> R[16][J*16] in LDS,
// then a per-edge CG contraction (2 lanes/edge, 2 output channels each).

__device__ __constant__ int c_wjstride[9] = {1, 3, 5, 3, 27, 45, 5, 45, 125};

template<bool IS_KEY>
__global__ __launch_bounds__(32 * WAVES)
void edge_radial_kernel(const int* __restrict__ u, const int* __restrict__ v,
    const float* __restrict__ f0, const float* __restrict__ f1,
    const float* __restrict__ f2, WjPtrs wjp,
    const float* __restrict__ ivg, const float* __restrict__ qg,
    const float* __restrict__ ag,
    const float* __restrict__ W1, const float* __restrict__ B1,
    const float* __restrict__ G1, const float* __restrict__ BE1,
    const float* __restrict__ W2, const float* __restrict__ B2,
    const float* __restrict__ G2, const float* __restrict__ BE2,
    const float* __restrict__ W3, const float* __restrict__ B3,
    float* __restrict__ dotg, float* __restrict__ msgg, int E)
{
  __shared__ float smem[WAVES * (2240 + WJMAX)];
  const int wid  = threadIdx.x >> 5;
  const int lane = threadIdx.x & 31;
  const int tile = blockIdx.x * WAVES + wid;
  const int e0   = tile * 16;
  if (e0 >= E) return;                      // wave-uniform

  float* IV = smem + wid * (2240 + WJMAX);  // [16][8]  (iv padded with zeros)
  float* H  = IV + 128;                     // [16][16] hidden
  float* R  = H  + 256;                     // [16][80] layer-3 output
  float* FV = R  + 1280;                    // [16][36] gathered node features
  float* WJ = FV + 576;                     // [16][<=125] staged wj tile

  // warm the wj stream for this tile (global_prefetch_b8)
  if (lane < 9)
    __builtin_prefetch(wjp.p[lane] + (size_t)e0 * c_wjstride[lane], 0, 0);

  // stage iv tile (128 entries = 4 per lane; exact multiple of 32)
  for (int idx = lane; idx < 128; idx += 32) {
    int e = idx >> 3, c = idx & 7;
    IV[idx] = (c < 5) ? ivg[(size_t)(e0 + e) * 5 + c] : 0.f;
  }
  // stage gathered features f[k][node]  (576 entries, 18 per lane)
  for (int idx = lane; idx < 16 * 36; idx += 32) {
    int e = idx / 36, t = idx % 36;
    int k, i, m;
    if (t < 4)       { k = 0; i = t;            m = 0;            }
    else if (t < 16) { k = 1; i = (t - 4) / 3;  m = (t - 4) % 3;  }
    else             { k = 2; i = (t - 16) / 5; m = (t - 16) % 5; }
    int node = IS_KEY ? v[e0 + e] : u[e0 + e];
    int nk = 2 * k + 1;
    const float* fk = (k == 0) ? f0 : ((k == 1) ? f1 : f2);
    FV[idx] = fk[((size_t)node * 4 + i) * nk + m];
  }

  const int elocal = lane >> 1, part = lane & 1;
  const int o0 = part * 2, o1 = o0 + 1;
  const int eg = e0 + elocal;
  float acc[2][9];
#pragma unroll
  for (int t = 0; t < 9; ++t) { acc[0][t] = 0.f; acc[1][t] = 0.f; }

  const int s = IS_KEY ? 1 : 0;
  for (int kd = 0; kd < 3; ++kd) {
    for (int ld = 0; ld < 3; ++ld) {
      const int J = 2 * ((kd < ld) ? kd : ld) + 1;
      const int p = s * 9 + kd * 3 + ld;
      const int nk = 2 * kd + 1, nl = 2 * ld + 1;
      const int stride = J * nl * nk;           // wj floats per edge
      const float* gwj = wjp.p[kd * 3 + ld] + (size_t)e0 * stride;

      // ---- stage wj tile into LDS; async copy overlaps the WMMAs below ----
#ifdef ASYNC_WJ
      for (int idx = lane; idx < 16 * stride; idx += 32)
        __builtin_amdgcn_global_load_async_to_lds_b32(
            (as1_i32p)(uintptr_t)(const void*)(gwj + idx),
            (as3_i32p)(WJ + idx), 0, 0);
#else
      for (int idx = lane; idx < 16 * stride; idx += 32) WJ[idx] = gwj[idx];
#endif

      // radial MLP: 5 -> 16 -> 16 -> J*16   (all on WMMA f32 16x16x4)
      layer16<5, true >(IV, 8, W1 + p * 80,  5, B1 + p * 16, G1 + p * 16,
                        BE1 + p * 16, H, 16, lane);
      layer16<16, true>(H, 16, W2 + p * 256, 16, B2 + p * 16, G2 + p * 16,
                        BE2 + p * 16, H, 16, lane);
      for (int j = 0; j < J; ++j)
        layer16<16, false>(H, 16, W3 + p * RST * 16 + j * 256, 16,
                           B3 + p * RST + j * 16, nullptr, nullptr,
                           R + j * 16, RST, lane);

#ifdef ASYNC_WJ
#if __has_builtin(__builtin_amdgcn_s_wait_asynccnt)
      __builtin_amdgcn_s_wait_asynccnt(0);
#else
      asm volatile("s_wait_asynccnt 0x0" ::: "memory");
#endif
#endif

      // CG contraction: out[o,l'] += sum_{j,i} Rm[j,o,i] *
      //                              (sum_{k'} wj[j,l',k'] * f[i,k'])
      const int lofs = ld * ld;
      const int kofs = (kd == 0) ? 0 : ((kd == 1) ? 4 : 16);
      const float* Re  = R  + elocal * RST;
      const float* fe  = FV + elocal * 36 + kofs;
      const float* wje = WJ + elocal * stride;
      for (int j = 0; j < J; ++j) {
        const float* wjr = wje + j * nl * nk;
        for (int i = 0; i < 4; ++i) {
          float r0 = Re[j * 16 + o0 * 4 + i];
          float r1 = Re[j * 16 + o1 * 4 + i];
          for (int lp = 0; lp < nl; ++lp) {
            float uu = 0.f;
            for (int kp = 0; kp < nk; ++kp)
              uu = fmaf(wjr[lp * nk + kp], fe[i * nk + kp], uu);
            acc[0][lofs + lp] = fmaf(r0, uu, acc[0][lofs + lp]);
            acc[1][lofs + lp] = fmaf(r1, uu, acc[1][lofs + lp]);
          }
        }
      }
    }
  }

  if (IS_KEY) {
    int vd = v[eg];
    float d = 0.f;
    for (int lp = 0; lp < 9; ++lp)
      d += qg[(size_t)vd * 36 + o0 * 9 + lp] * acc[0][lp] +
           qg[(size_t)vd * 36 + o1 * 9 + lp] * acc[1][lp];
    d = pair_add(d);
    if (part == 0) dotg[eg] = d;
  } else {
    float ae = ag[eg];
    for (int lp = 0; lp < 9; ++lp) {
      msgg[(size_t)eg * 36 + o0 * 9 + lp] = ae * acc[0][lp];
      msgg[(size_t)eg * 36 + o1 * 9 + lp] = ae * acc[1][lp];
    }
  }
}

// ---------------- self-interaction (per degree, 16-node tiles) ----------------

__global__ __launch_bounds__(32 * WAVES)
void self_interact_kernel(const float* __restrict__ f0,
    const float* __restrict__ f1, const float* __restrict__ f2,
    const float* __restrict__ sW1, const float* __restrict__ sb1,
    const float* __restrict__ sg1, const float* __restrict__ sbe1,
    const float* __restrict__ sW2, const float* __restrict__ sb2,
    const float* __restrict__ sg2, const float* __restrict__ sbe2,
    const float* __restrict__ sW3, const float* __restrict__ sb3,
    float* __restrict__ out, int N)
{
  __shared__ float smem[WAVES * 1088];
  const int wid  = threadIdx.x >> 5;
  const int lane = threadIdx.x & 31;
  const int ntiles = N >> 4;
  const int gw = blockIdx.x * WAVES + wid;
  if (gw >= 3 * ntiles) return;             // wave-uniform
  const int ld = gw / ntiles;
  const int nb = (gw % ntiles) * 16;

  float* FT = smem + wid * 1088;            // [16][4*nl]   (nl<=5 -> 320)
  float* IN = FT + 320;                     // [16][16] inner products
  float* H  = IN + 256;                     // [16][16] hidden
  float* SW = H  + 256;                     // [16][16] siw

  const float* fptr = (ld == 0) ? f0 : ((ld == 1) ? f1 : f2);
  const int nl = 2 * ld + 1, per = 4 * nl;  // per in {4,12,20} -> counts %32==0

  for (int idx = lane; idx < 16 * per; idx += 32)
    FT[idx] = fptr[(size_t)nb * per + idx];

  for (int idx = lane; idx < 256; idx += 32) {
    int nd = idx >> 4, cd = idx & 15, cc = cd >> 2, dd = cd & 3;
    float s = 0.f;
    for (int m = 0; m < nl; ++m)
      s = fmaf(FT[nd * per + cc * nl + m], FT[nd * per + dd * nl + m], s);
    IN[idx] = s;
  }

  layer16<16, true >(IN, 16, sW1 + ld * 256, 16, sb1 + ld * 16, sg1 + ld * 16,
                     sbe1 + ld * 16, H, 16, lane);
  layer16<16, true >(H, 16, sW2 + ld * 256, 16, sb2 + ld * 16, sg2 + ld * 16,
                     sbe2 + ld * 16, H, 16, lane);
  layer16<16, false>(H, 16, sW3 + ld * 256, 16, sb3 + ld * 16, nullptr, nullptr,
                     SW, 16, lane);

  const int lofs = ld * ld;
  for (int idx = lane; idx < 16 * per; idx += 32) {
    int nd = idx / per, t = idx % per, o = t / nl, m = t % nl;
    float s = 0.f;
    for (int i = 0; i < 4; ++i)
      s = fmaf(SW[nd * 16 + o * 4 + i], FT[nd * per + i * nl + m], s);
    out[(size_t)(nb + nd) * 36 + o * 9 + lofs + m] += s;
  }
}

// ---------------- launcher ----------------

extern "C" void kernel_launch(void* const* d_in, const int* in_sizes, int n_in,
                              void* d_out, int out_size, void* d_ws, size_t ws_size,
                              hipStream_t stream) {
  const int*   u    = (const int*)d_in[0];
  const int*   v    = (const int*)d_in[1];
  const float* dist = (const float*)d_in[2];
  const float* f0   = (const float*)d_in[3];
  const float* f1   = (const float*)d_in[4];
  const float* f2   = (const float*)d_in[5];
  WjPtrs wjp;
  for (int i = 0; i < 9; ++i) wjp.p[i] = (const float*)d_in[6 + i];
  const float* wq  = (const float*)d_in[15];
  const float* rW1 = (const float*)d_in[16], *rb1 = (const float*)d_in[17];
  const float* rg1 = (const float*)d_in[18], *rbe1 = (const float*)d_in[19];
  const float* rW2 = (const float*)d_in[20], *rb2 = (const float*)d_in[21];
  const float* rg2 = (const float*)d_in[22], *rbe2 = (const float*)d_in[23];
  const float* rW3 = (const float*)d_in[24], *rb3 = (const float*)d_in[25];
  const float* sW1 = (const float*)d_in[26], *sb1 = (const float*)d_in[27];
  const float* sg1 = (const float*)d_in[28], *sbe1 = (const float*)d_in[29];
  const float* sW2 = (const float*)d_in[30], *sb2 = (const float*)d_in[31];
  const float* sg2 = (const float*)d_in[32], *sbe2 = (const float*)d_in[33];
  const float* sW3 = (const float*)d_in[34], *sb3 = (const float*)d_in[35];

  const int E   = in_sizes[0];
  const int N   = in_sizes[3] / 4;
  const int deg = E / N;

  float* ws   = (float*)d_ws;               // ~9.1 MB used
  float* ivg  = ws;                         // E*5
  float* qg   = ivg  + (size_t)E * 5;       // N*36
  float* dotg = qg   + (size_t)N * 36;      // E
  float* ag   = dotg + E;                   // E
  float* msg  = ag   + E;                   // E*36
  float* outf = (float*)d_out;

  q_kernel <<<(N * 36 + 255) / 256, 256, 0, stream>>>(wq, f0, f1, f2, qg, N);
  iv_kernel<<<(E + 255) / 256, 256, 0, stream>>>(u, v, dist, f0, ivg, E);

  const int tiles = E / 16;
  const int eb = (tiles + WAVES - 1) / WAVES;
  edge_radial_kernel<true><<<eb, 32 * WAVES, 0, stream>>>(u, v, f0, f1, f2, wjp,
      ivg, qg, ag, rW1, rb1, rg1, rbe1, rW2, rb2, rg2, rbe2, rW3, rb3,
      dotg, msg, E);
  softmax_kernel<<<(N + 255) / 256, 256, 0, stream>>>(dotg, ag, N, deg);
  edge_radial_kernel<false><<<eb, 32 * WAVES, 0, stream>>>(u, v, f0, f1, f2, wjp,
      ivg, qg, ag, rW1, rb1, rg1, rbe1, rW2, rb2, rg2, rbe2, rW3, rb3,
      dotg, msg, E);
  reduce_kernel<<<(N * 36 + 255) / 256, 256, 0, stream>>>(msg, outf, N, deg);

  const int siwv = 3 * (N / 16);
  self_interact_kernel<<<(siwv + WAVES - 1) / WAVES, 32 * WAVES, 0, stream>>>(
      f0, f1, f2, sW1, sb1, sg1, sbe1, sW2, sb2, sg2, sbe2, sW3, sb3, outf, N);
}